// FaceDetector_34007551050117
// MI455X (gfx1250) — compile-verified
//
#include <hip/hip_runtime.h>
#include <math.h>

// ---------------- problem constants (from reference) ----------------
#define NB      8          // batch
#define NP      60160      // priors
#define TOPK    2048
#define KEEPK   512
#define CONF_T  0.3f
#define NMS_T   0.3f
#define VAR0    0.1f
#define VAR1    0.2f

typedef __attribute__((ext_vector_type(2))) float v2f;
typedef __attribute__((ext_vector_type(8))) float v8f;
typedef int v4i __attribute__((vector_size(16)));   // matches async builtin param type
typedef __attribute__((ext_vector_type(4))) unsigned v4u_;
typedef __attribute__((ext_vector_type(4))) int v4s_;
typedef __attribute__((ext_vector_type(8))) int v8s_;
typedef unsigned long long u64;
typedef unsigned int u32;
typedef unsigned short u16;

#define GLOBAL_AS __attribute__((address_space(1)))
#define LDS_AS    __attribute__((address_space(3)))

#if defined(__gfx1250__)
#if __has_builtin(__builtin_amdgcn_wmma_f32_16x16x4_f32)
#define HAVE_WMMA_F32X4 1
#endif
#if __has_builtin(__builtin_amdgcn_global_load_async_to_lds_b128) && \
    __has_builtin(__builtin_amdgcn_global_load_async_to_lds_b32)
#define HAVE_ASYNC_LDS 1
#endif
#if __has_builtin(__builtin_amdgcn_tensor_load_to_lds)
#define HAVE_TDM 1
#endif
#endif

#if defined(HAVE_ASYNC_LDS)
__device__ __forceinline__ void async_g2l_b128(void* lds, const void* g) {
    __builtin_amdgcn_global_load_async_to_lds_b128(
        (GLOBAL_AS v4i*)(void*)g, (LDS_AS v4i*)lds, 0, 0);
}
__device__ __forceinline__ void async_g2l_b32(void* lds, const void* g) {
    __builtin_amdgcn_global_load_async_to_lds_b32(
        (GLOBAL_AS int*)(void*)g, (LDS_AS int*)lds, 0, 0);
}
#if __has_builtin(__builtin_amdgcn_s_wait_asynccnt)
#define WAIT_ASYNC0() __builtin_amdgcn_s_wait_asynccnt(0)
#define WAIT_ASYNC1() __builtin_amdgcn_s_wait_asynccnt(1)
#else
#define WAIT_ASYNC0() asm volatile("s_wait_asynccnt 0x0" ::: "memory")
#define WAIT_ASYNC1() asm volatile("s_wait_asynccnt 0x1" ::: "memory")
#endif
#endif

#if defined(HAVE_TDM)
#if __has_builtin(__builtin_amdgcn_s_wait_tensorcnt)
#define WAIT_TENSOR0() __builtin_amdgcn_s_wait_tensorcnt(0)
#define WAIT_TENSOR1() __builtin_amdgcn_s_wait_tensorcnt(1)
#else
#define WAIT_TENSOR0() asm volatile("s_wait_tensorcnt 0x0" ::: "memory")
#define WAIT_TENSOR1() asm volatile("s_wait_tensorcnt 0x1" ::: "memory")
#endif

__device__ __forceinline__ u32 lds_offset(const void* p) {
    return (u32)(unsigned long long)(LDS_AS void*)(void*)p;   // AS3 ptr value == LDS byte offset
}

// 1-D TDM tile load: nelem0 = tensor length (8B units, OOB reads return 0),
// tile0 = tile length (8B units). D# bitfields per CDNA5 ISA ch.8.
__device__ __forceinline__ void tdm_load_1d(u32 ldsoff, const void* gaddr, u32 nelem0, u32 tile0) {
    u64 ga = (u64)(unsigned long long)gaddr;
    v4u_ g0;
    g0[0] = 1u;                                               // count=1 (valid), user mode
    g0[1] = ldsoff;                                           // lds_addr (bytes)
    g0[2] = (u32)ga;                                          // global_addr[31:0]
    g0[3] = (u32)((ga >> 32) & 0x01FFFFFFu) | 0x80000000u;    // global_addr[56:32] | type=2
    v8s_ g1;
    g1[0] = (int)(3u << 16);                                  // data_size=3 (8B); wg_mask=0; no flags
    g1[1] = (int)((nelem0 & 0xFFFFu) << 16);                  // tensor_dim0[15:0] @ bits[63:48]
    g1[2] = (int)(((nelem0 >> 16) & 0xFFFFu) | (1u << 16));   // tensor_dim0[31:16]; tensor_dim1=1
    g1[3] = (int)((tile0 & 0xFFFFu) << 16);                   // tile_dim0 @ bits[127:112]
    g1[4] = 0;                                                // tile_dim1=0 (unused), tile_dim2=0
    g1[5] = (int)nelem0;                                      // tensor_dim0_stride[31:0]
    g1[6] = 0;                                                // stride0[47:32]=0; dim1_stride[15:0]=0
    g1[7] = 0;
    v4s_ z4 = {0, 0, 0, 0};
#if __clang_major__ >= 23
    v8s_ z8 = {0, 0, 0, 0, 0, 0, 0, 0};
    __builtin_amdgcn_tensor_load_to_lds(g0, g1, z4, z4, z8, 0);   // 6-arg (clang-23 lane)
#else
    __builtin_amdgcn_tensor_load_to_lds(g0, g1, z4, z4, 0);       // 5-arg (ROCm 7.2 lane)
#endif
}
#endif

// ---------------- kernel 1: priors ----------------
__global__ void k_priors(const int* __restrict__ hptr, const int* __restrict__ wptr,
                         float* __restrict__ priors) {
    int p = blockIdx.x * blockDim.x + threadIdx.x;
    if (p >= NP) return;
    const int h = *hptr, w = *wptr;
    const int steps[4] = {8, 16, 32, 64};
    const int nmin[4]  = {3, 2, 2, 3};
    const float msz[4][3] = {{10.f,16.f,24.f},{32.f,48.f,0.f},{64.f,96.f,0.f},{128.f,192.f,256.f}};
    int rem = p;
    for (int k = 0; k < 4; ++k) {
        const int step = steps[k];
        const int fh = (h + step - 1) / step;
        const int fw = (w + step - 1) / step;
        const int M  = nmin[k];
        const int cnt = fh * fw * M;
        if (rem < cnt) {
            int i  = rem / (fw * M);
            int r2 = rem % (fw * M);
            int j  = r2 / M;
            int m  = r2 % M;
            float cx  = (j + 0.5f) * (float)step / (float)w;
            float cy  = (i + 0.5f) * (float)step / (float)h;
            float skx = msz[k][m] / (float)w;
            float sky = msz[k][m] / (float)h;
            reinterpret_cast<float4*>(priors)[p] = make_float4(cx, cy, skx, sky);
            return;
        }
        rem -= cnt;
    }
}

// ---------------- kernel 2: scores -> sortable 64-bit keys ----------------
__global__ void k_score(const float* __restrict__ conf, const float* __restrict__ iouin,
                        u64* __restrict__ keys) {
    int idx = blockIdx.x * blockDim.x + threadIdx.x;
    if (idx >= NB * NP) return;
    int p = idx % NP;
    __builtin_prefetch(conf + (size_t)idx * 2 + 4096, 0, 0);   // -> global_prefetch_b8
    float c1 = conf[(size_t)idx * 2 + 1];
    float io = iouin[idx];
    io = fminf(fmaxf(io, 0.0f), 1.0f);
    float s = sqrtf(c1 * io);
    float masked = (s > CONF_T) ? s : -1.0f;
    u32 bits = __float_as_uint(masked);
    u32 key32 = (bits & 0x80000000u) ? ~bits : (bits | 0x80000000u);
    keys[idx] = ((u64)key32 << 32) | (u32)(~(u32)p);
}

// ---------------- kernel 3: per-batch exact top-2048 via LDS bitonic ----------------
__device__ __forceinline__ void bitonic_sort_desc(u64* a, int tid) {
    for (int k = 2; k <= TOPK; k <<= 1) {
        for (int j = k >> 1; j > 0; j >>= 1) {
            __syncthreads();
            int t = tid;                  // 1024 threads == TOPK/2 pairs
            int i  = ((t & ~(j - 1)) << 1) | (t & (j - 1));
            int pp = i | j;
            bool desc = ((i & k) == 0);
            u64 x = a[i], y = a[pp];
            bool sw = desc ? (x < y) : (x > y);
            if (sw) { a[i] = y; a[pp] = x; }
        }
    }
    __syncthreads();
}

__device__ __forceinline__ void bitonic_merge_desc(u64* a, int tid) {
    for (int j = TOPK >> 1; j > 0; j >>= 1) {
        __syncthreads();
        int t = tid;
        int i  = ((t & ~(j - 1)) << 1) | (t & (j - 1));
        int pp = i | j;
        u64 x = a[i], y = a[pp];
        if (x < y) { a[i] = y; a[pp] = x; }
    }
    __syncthreads();
}

// Chunk stream: TDM 1-D tile per chunk (preferred) or per-lane async b128, double-buffered
// so the next chunk's HBM latency hides behind the bitonic sort of the current chunk.
__global__ __launch_bounds__(1024) void k_topk(const u64* __restrict__ keys,
                                               float* __restrict__ valtop,
                                               int* __restrict__ idxtop) {
    __shared__ u64 best[TOPK];
    __shared__ u64 chnk[2][TOPK];
    const int b = blockIdx.x;
    const int tid = threadIdx.x;
    const u64* kb = keys + (size_t)b * NP;
    const int nch = (NP - TOPK + TOPK - 1) / TOPK;   // 29 remaining chunks

#if defined(HAVE_ASYNC_LDS)
    async_g2l_b128(&best[2 * tid], kb + 2 * tid);     // first TOPK keys -> best
    WAIT_ASYNC0();
    __syncthreads();
#else
    for (int t = tid; t < TOPK; t += 1024) best[t] = kb[t];
#endif
    bitonic_sort_desc(best, tid);

#if defined(HAVE_TDM)
    if (tid < 32)                                     // wave 0 drives the TDM pipeline
        tdm_load_1d(lds_offset(&chnk[0][0]), kb + TOPK, (u32)(NP - TOPK), TOPK);
#elif defined(HAVE_ASYNC_LDS)
    {
        size_t g = (size_t)TOPK + 2 * tid;
        async_g2l_b128(&chnk[0][2 * tid], kb + ((g + 1 < NP) ? g : (NP - 2)));
    }
#endif
    for (int ci = 0; ci < nch; ++ci) {
        const int cur = ci & 1;
        const int base = TOPK + ci * TOPK;
#if defined(HAVE_TDM)
        if (tid < 32) {
            if (ci + 1 < nch) {
                int nbase = TOPK + (ci + 1) * TOPK;
                tdm_load_1d(lds_offset(&chnk[cur ^ 1][0]), kb + nbase, (u32)(NP - nbase), TOPK);
                WAIT_TENSOR1();                       // in-order: current tile complete
            } else {
                WAIT_TENSOR0();
            }
        }
        __syncthreads();                              // publish TDM-written LDS to all waves
        for (int t = tid; t < TOPK; t += 1024)
            if (base + t >= NP) chnk[cur][t] = 0ull;  // tail padding (< any real key)
#elif defined(HAVE_ASYNC_LDS)
        if (ci + 1 < nch) {
            size_t g = (size_t)TOPK + (size_t)(ci + 1) * TOPK + 2 * tid;
            async_g2l_b128(&chnk[cur ^ 1][2 * tid], kb + ((g + 1 < NP) ? g : (NP - 2)));
            WAIT_ASYNC1();
        } else {
            WAIT_ASYNC0();
        }
        __syncthreads();
        for (int t = tid; t < TOPK; t += 1024)
            if (base + t >= NP) chnk[cur][t] = 0ull;
#else
        for (int t = tid; t < TOPK; t += 1024) {
            int g = base + t;
            chnk[cur][t] = (g < NP) ? kb[g] : 0ull;
        }
#endif
        bitonic_sort_desc(chnk[cur], tid);
        for (int t = tid; t < TOPK; t += 1024) {
            u64 x = best[t], y = chnk[cur][TOPK - 1 - t];
            best[t] = (x > y) ? x : y;       // bitonic top-K merge trick
        }
        bitonic_merge_desc(best, tid);       // trailing barrier frees chnk[cur]
    }

    for (int t = tid; t < TOPK; t += 1024) {
        u64 k = best[t];
        u32 s = (u32)(k >> 32);
        u32 fb = (s & 0x80000000u) ? (s & 0x7fffffffu) : ~s;
        valtop[b * TOPK + t] = __uint_as_float(fb);
        idxtop[b * TOPK + t] = (int)(~(u32)k);
    }
}

// ---------------- kernel 4: gather + decode survivors ----------------
__global__ void k_gather_decode(const float* __restrict__ loc, const float* __restrict__ priors,
                                const int* __restrict__ idxtop,
                                const int* __restrict__ hptr, const int* __restrict__ wptr,
                                float* __restrict__ boxes14, float* __restrict__ boxes4,
                                float* __restrict__ area) {
    int t = blockIdx.x * blockDim.x + threadIdx.x;
    if (t >= NB * TOPK) return;
    int b = t / TOPK;
    int p = idxtop[t];
    if ((unsigned)p >= (unsigned)NP) p = 0;   // defensive
    float W = (float)*wptr, H = (float)*hptr;
    const float* L = loc + ((size_t)b * NP + p) * 14;
    float4 pr = reinterpret_cast<const float4*>(priors)[p];
    float cx = pr.x + L[0] * VAR0 * pr.z;
    float cy = pr.y + L[1] * VAR0 * pr.w;
    float bw = pr.z * __expf(L[2] * VAR0);
    float bh = pr.w * __expf(L[3] * VAR1);
    float x1 = (cx - bw * 0.5f) * W;
    float y1 = (cy - bh * 0.5f) * H;
    float x2 = (cx + bw * 0.5f) * W;
    float y2 = (cy + bh * 0.5f) * H;
    float* O = boxes14 + (size_t)t * 14;
    O[0] = x1; O[1] = y1; O[2] = x2; O[3] = y2;
#pragma unroll
    for (int k = 0; k < 5; ++k) {
        O[4 + 2 * k] = (pr.x + L[4 + 2 * k] * VAR0 * pr.z) * W;
        O[5 + 2 * k] = (pr.y + L[5 + 2 * k] * VAR0 * pr.w) * H;
    }
    reinterpret_cast<float4*>(boxes4)[t] = make_float4(x1, y1, x2, y2);  // 16B-aligned for IoU
    area[t] = (x2 - x1) * (y2 - y1);
}

// ---------------- kernel 5: IoU tiles; WMMA computes area_i + area_j ----------------
// One wave32 per (row-tile, 4 col-tiles). A = [area_i | 1 | 0 | 0] (16x4),
// B = [1 ; area_j ; 0 ; 0] (4x16) => D[M][N] = area_i[M] + area_j[N] per V_WMMA_F32_16X16X4_F32.
// Threshold without division: iou > t  <=>  inter*(1+t) > t*(area_i+area_j).
__global__ __launch_bounds__(32) void k_iou_bits(const float* __restrict__ boxes4,
                                                 const float* __restrict__ area,
                                                 u16* __restrict__ supbits) {
    const int b   = blockIdx.z;
    const int ti  = blockIdx.y;            // row tile
    const int tj0 = blockIdx.x * 4;        // first of 4 col tiles
    const int lane = threadIdx.x;
    const int N = lane & 15;
    const int g = lane >> 4;

    const float* B4 = boxes4 + (size_t)b * TOPK * 4;
    const float* AR = area   + (size_t)b * TOPK;

    __shared__ float4 rowbox[16];
    __shared__ float  rowarea[16];
    __shared__ u32    rowbits[16];

#if defined(HAVE_ASYNC_LDS)
    if (lane < 16) async_g2l_b128(&rowbox[lane], B4 + (size_t)(ti * 16 + lane) * 4);
    else           async_g2l_b32(&rowarea[lane - 16], AR + ti * 16 + (lane - 16));
    WAIT_ASYNC0();
#else
    if (lane < 16) rowbox[lane] = reinterpret_cast<const float4*>(B4)[ti * 16 + lane];
    else           rowarea[lane - 16] = AR[ti * 16 + (lane - 16)];
#endif
    __syncthreads();                        // single-wave WG: wave barrier
    const float rarea = (lane < 16) ? rowarea[lane] : 0.0f;

#pragma unroll
    for (int u = 0; u < 4; ++u) {
        const int tj = tj0 + u;
        float sums[8];
#if defined(HAVE_WMMA_F32X4)
        v2f av = {0.0f, 0.0f};
        v2f bv = {0.0f, 0.0f};
        if (lane < 16) {
            av[0] = rarea; av[1] = 1.0f;                 // A cols K=0 (area_i), K=1 (ones)
            bv[0] = 1.0f;  bv[1] = AR[tj * 16 + lane];   // B rows K=0 (ones), K=1 (area_j)
        }
        v8f c = {0.f,0.f,0.f,0.f,0.f,0.f,0.f,0.f};
        c = __builtin_amdgcn_wmma_f32_16x16x4_f32(false, av, false, bv, (short)0, c, false, false);
#pragma unroll
        for (int r = 0; r < 8; ++r) sums[r] = c[r];      // lane holds rows M = r + 8*g, col N
#else
        float carea = AR[tj * 16 + N];
#pragma unroll
        for (int r = 0; r < 8; ++r) sums[r] = rowarea[g * 8 + r] + carea;
#endif
        float4 cb = reinterpret_cast<const float4*>(B4)[tj * 16 + N];

        if (lane < 16) rowbits[lane] = 0u;
        __syncthreads();
#pragma unroll
        for (int r = 0; r < 8; ++r) {
            const int M = g * 8 + r;
            float4 rb = rowbox[M];
            float iw = fminf(rb.z, cb.z) - fmaxf(rb.x, cb.x);
            float ih = fminf(rb.w, cb.w) - fmaxf(rb.y, cb.y);
            float inter = fmaxf(iw, 0.0f) * fmaxf(ih, 0.0f);
            if (inter * (1.0f + NMS_T) > NMS_T * sums[r])   // divide-free IoU > thr
                atomicOr(&rowbits[M], 1u << N);
        }
        __syncthreads();
        if (lane < 16)
            supbits[((size_t)b * TOPK + ti * 16 + lane) * (TOPK / 16) + tj] = (u16)rowbits[lane];
        __syncthreads();
    }
}

// ---------------- kernel 6: greedy NMS on bitmasks ----------------
__global__ __launch_bounds__(64) void k_nms(const float* __restrict__ valtop,
                                            const u32* __restrict__ suprows,
                                            u32* __restrict__ keepbits) {
    const int b = blockIdx.x;
    const int t = threadIdx.x;              // 64 words x 32 bits = 2048 flags
    __shared__ u32 keepw[TOPK / 32];

    u32 wbits = 0u;
    for (int k = 0; k < 32; ++k) {
        float v = valtop[b * TOPK + t * 32 + k];
        if (v > CONF_T) wbits |= (1u << k);
    }
    keepw[t] = wbits;
    __syncthreads();

    for (int i = 0; i < TOPK; ++i) {
        const int wi = i >> 5, bi = i & 31;
        const bool keep_i = (keepw[wi] >> bi) & 1u;
        if (keep_i) {
            u32 m = suprows[((size_t)b * TOPK + i) * (TOPK / 32) + t];
            if (t < wi) m = 0u;
            else if (t == wi) {
                u32 low = (bi == 31) ? 0xFFFFFFFFu : ((1u << (bi + 1)) - 1u);
                m &= ~low;                  // only suppress columns j > i
            }
            keepw[t] &= ~m;
        }
        __syncthreads();
    }
    keepbits[b * (TOPK / 32) + t] = keepw[t];
}

// ---------------- kernel 7: stable partition + emit [512,15] ----------------
__global__ __launch_bounds__(256) void k_emit(const float* __restrict__ boxes14,
                                              const float* __restrict__ valtop,
                                              const u32* __restrict__ keepbits,
                                              float* __restrict__ out) {
    const int b = blockIdx.x;
    const int tid = threadIdx.x;
    __shared__ u32 kw[TOPK / 32];
    __shared__ int scan[TOPK / 32 + 1];

    if (tid < TOPK / 32) kw[tid] = keepbits[b * (TOPK / 32) + tid];
    __syncthreads();
    if (tid == 0) {
        int s = 0;
        for (int w = 0; w < TOPK / 32; ++w) { scan[w] = s; s += __popc(kw[w]); }
        scan[TOPK / 32] = s;
    }
    __syncthreads();
    const int total = scan[TOPK / 32];

    for (int i = tid; i < TOPK; i += 256) {
        int w = i >> 5, bit = i & 31;
        u32 word = kw[w];
        int kept = (int)((word >> bit) & 1u);
        int before = scan[w] + __popc(word & ((1u << bit) - 1u));
        int pos = kept ? before : (total + (i - before));
        if (pos < KEEPK) {
            float* O = out + ((size_t)b * KEEPK + pos) * 15;
            if (kept) {
                const float* S = boxes14 + ((size_t)b * TOPK + i) * 14;
#pragma unroll
                for (int q = 0; q < 14; ++q) O[q] = S[q];
                O[14] = valtop[b * TOPK + i];
            } else {
#pragma unroll
                for (int q = 0; q < 15; ++q) O[q] = 0.0f;
            }
        }
    }
}

// ---------------- host launcher ----------------
static inline size_t alignup(size_t x) { return (x + 255) & ~(size_t)255; }

extern "C" void kernel_launch(void* const* d_in, const int* in_sizes, int n_in,
                              void* d_out, int out_size, void* d_ws, size_t ws_size,
                              hipStream_t stream) {
    const float* loc   = (const float*)d_in[0];   // [B,P,14]
    const float* conf  = (const float*)d_in[1];   // [B,P,2]
    const float* iouin = (const float*)d_in[2];   // [B,P,1]
    const int*   hptr  = (const int*)d_in[3];     // scalar
    const int*   wptr  = (const int*)d_in[4];     // scalar
    float* out = (float*)d_out;                   // [B,512,15]

    char* w = (char*)d_ws;
    float* priors  = (float*)w;  w += alignup((size_t)NP * 4 * sizeof(float));
    u64*   keys    = (u64*)w;    w += alignup((size_t)NB * NP * sizeof(u64));
    float* valtop  = (float*)w;  w += alignup((size_t)NB * TOPK * sizeof(float));
    int*   idxtop  = (int*)w;    w += alignup((size_t)NB * TOPK * sizeof(int));
    float* boxes14 = (float*)w;  w += alignup((size_t)NB * TOPK * 14 * sizeof(float));
    float* boxes4  = (float*)w;  w += alignup((size_t)NB * TOPK * 4 * sizeof(float));
    float* area    = (float*)w;  w += alignup((size_t)NB * TOPK * sizeof(float));
    u16*   supbits = (u16*)w;    w += alignup((size_t)NB * TOPK * (TOPK / 16) * sizeof(u16));
    u32*   keepbit = (u32*)w;    w += alignup((size_t)NB * (TOPK / 32) * sizeof(u32));
    (void)ws_size; (void)n_in; (void)in_sizes; (void)out_size;

    k_priors<<<(NP + 255) / 256, 256, 0, stream>>>(hptr, wptr, priors);
    k_score<<<(NB * NP + 255) / 256, 256, 0, stream>>>(conf, iouin, keys);
    k_topk<<<NB, 1024, 0, stream>>>(keys, valtop, idxtop);
    k_gather_decode<<<(NB * TOPK + 255) / 256, 256, 0, stream>>>(loc, priors, idxtop, hptr, wptr,
                                                                 boxes14, boxes4, area);
    k_iou_bits<<<dim3(TOPK / 64, TOPK / 16, NB), 32, 0, stream>>>(boxes4, area, supbits);
    k_nms<<<NB, 64, 0, stream>>>(valtop, (const u32*)supbits, keepbit);
    k_emit<<<NB, 256, 0, stream>>>(boxes14, valtop, keepbit, out);
}